// StepNet_11785390260311
// MI455X (gfx1250) — compile-verified
//
#include <hip/hip_runtime.h>
#include <stdint.h>

// out[i] = values[ #{j : x[i] > breakpoints[j]} ]
// (exactly equivalent to the reference's one-hot-regions @ values for sorted
//  breakpoints, which linspace guarantees)

#define NBP 2048

typedef __attribute__((ext_vector_type(4))) unsigned int u32x4;
typedef __attribute__((ext_vector_type(8))) int         i32x8;
typedef __attribute__((ext_vector_type(4))) int         i32x4;

// Issue one TDM DMA: nelem consecutive f32 from global -> LDS byte address.
// D# layout per CDNA5 ISA (normal mode, 1-row tile).
__device__ __forceinline__ void tdm_load_f32_1d(uint32_t lds_byte_addr,
                                                const void* gptr,
                                                uint32_t nelem) {
  const uint64_t ga = (uint64_t)(uintptr_t)gptr;

  u32x4 g0;
  g0[0] = 1u;                                     // [1:0] count=1 (valid), user mode
  g0[1] = lds_byte_addr;                          // [63:32] lds_addr (bytes)
  g0[2] = (uint32_t)(ga & 0xFFFFFFFFu);           // [95:64] global_addr lo
  g0[3] = (uint32_t)((ga >> 32) & 0x01FFFFFFu)    // [120:96] global_addr[56:32]
        | (2u << 30);                             // [127:126] type = 2 ("image")

  i32x8 g1;
  g1[0] = (int)(2u << 16);                        // wg_mask=0, data_size=2 (4B)
  g1[1] = (int)((nelem & 0xFFFFu) << 16);         // tensor_dim0[15:0] @ [63:48]
  g1[2] = (int)(((nelem >> 16) & 0xFFFFu)         // tensor_dim0[31:16] @ [79:64]
        | (1u << 16));                            // tensor_dim1 = 1 @ [111:80]
  g1[3] = (int)((nelem & 0xFFFFu) << 16);         // tile_dim0 = nelem @ [127:112]
  g1[4] = 1;                                      // tile_dim1=1, tile_dim2=0(unused)
  g1[5] = (int)nelem;                             // tensor_dim0_stride lo
  g1[6] = 0;                                      // stride0 hi, stride1 lo
  g1[7] = 0;                                      // stride1 hi

  i32x4 g2;                                       // normal mode (iterate=0)
  g2[0] = 1;                                      // tensor_dim2 = 1
  g2[1] = 1;                                      // tensor_dim3 = 1
  g2[2] = 0;                                      // tensor_dim2_stride lo
  g2[3] = 0;                                      // stride2 hi, tile_dim3=0(unused)

  i32x4 g3;
  g3[0] = 0;                                      // tensor_dim3_stride lo
  g3[1] = (int)(1u << 16);                        // tensor_dim4 = 1 @ [79:48]
  g3[2] = 0;                                      // dim4 hi, tile_dim4=0(unused)
  g3[3] = 0;                                      // reserved

  const i32x8 g4 = {0, 0, 0, 0, 0, 0, 0, 0};      // 6-arg form: reserved group
  __builtin_amdgcn_tensor_load_to_lds(g0, g1, g2, g3, g4, /*cpol=*/0);
}

__global__ __launch_bounds__(256) void stepnet_lut_kernel(
    const float* __restrict__ x,
    const float* __restrict__ bp,
    const float* __restrict__ val,
    float* __restrict__ out,
    int nvec4) {
  __shared__ __align__(16) float s_bp[NBP];
  __shared__ __align__(16) float s_val[NBP + 4];   // 2049 used, padded

  // Wave 0 of the workgroup issues both tensor DMAs and waits on TENSORcnt.
  if ((threadIdx.x >> 5) == 0) {
    tdm_load_f32_1d((uint32_t)(uintptr_t)&s_bp[0], bp, NBP);
    tdm_load_f32_1d((uint32_t)(uintptr_t)&s_val[0], val, NBP + 1);
    __builtin_amdgcn_s_wait_tensorcnt(0);
  }
  __syncthreads();

  const int i4 = blockIdx.x * blockDim.x + threadIdx.x;
  if (i4 >= nvec4) return;

  const float4 xv = ((const float4*)x)[i4];        // coalesced b128 load
  const float xs[4] = {xv.x, xv.y, xv.z, xv.w};
  float r[4];

  #pragma unroll
  for (int e = 0; e < 4; ++e) {
    const float v = xs[e];
    // Branchless lower-bound over the sorted LDS table: result in [0, NBP].
    int pos = 0;
    #pragma unroll
    for (int s = NBP >> 1; s > 0; s >>= 1)
      pos += (v > s_bp[pos + s - 1]) ? s : 0;
    pos += (v > s_bp[pos]) ? 1 : 0;                // pos == count(x > bp[j])
    r[e] = s_val[pos];                             // exact gather
  }

  ((float4*)out)[i4] = make_float4(r[0], r[1], r[2], r[3]);
}

extern "C" void kernel_launch(void* const* d_in, const int* in_sizes, int n_in,
                              void* d_out, int out_size, void* d_ws, size_t ws_size,
                              hipStream_t stream) {
  const float* x   = (const float*)d_in[0];   // (65536,1) f32
  const float* bp  = (const float*)d_in[1];   // (2048,)   f32 sorted
  const float* val = (const float*)d_in[2];   // (2049,1)  f32
  float* out = (float*)d_out;                 // (65536,1) f32

  const int n = in_sizes[0];                  // 65536 (multiple of 1024)
  const int nvec4 = n / 4;
  const int threads = 256;
  const int blocks = (nvec4 + threads - 1) / threads;

  hipLaunchKernelGGL(stepnet_lut_kernel, dim3(blocks), dim3(threads), 0, stream,
                     x, bp, val, out, nvec4);
}